// crysAE_53145925321464
// MI455X (gfx1250) — compile-verified
//
#include <hip/hip_runtime.h>
#include <hip/hip_bf16.h>
#include <math.h>

// ---------------- problem constants ----------------
#define B_G   1024
#define N_A   20
#define D_H   128
#define K_E   64
#define A_DIM 92
#define SG_N  230

// output segment offsets (floats) in d_out, reference return order:
// (edge_probs, adjs), atom_feat, sg_pred, z_G, z
#define EP_OFF   ((size_t)0)
#define ADJ_OFF  ((size_t)26214400)   // B*N*N*K
#define AF_OFF   ((size_t)26624000)   // + B*N*N
#define SGP_OFF  ((size_t)28508160)   // + B*N*A_DIM
#define ZG_OFF   ((size_t)28743680)   // + B*SG
#define ZO_OFF   ((size_t)28874752)   // + B*D

typedef __attribute__((ext_vector_type(2))) float v2f;
typedef __attribute__((ext_vector_type(8))) float v8f;

__device__ __forceinline__ v8f wmma4(v2f a, v2f b, v8f c) {
  // D(16x16,f32) = A(16x4,f32) x B(4x16,f32) + C
  return __builtin_amdgcn_wmma_f32_16x16x4_f32(false, a, false, b, (short)0, c,
                                               false, false);
}

// ---------------- pool (z_G) + z passthrough ----------------
__global__ void pool_copy_kernel(const float* __restrict__ z,
                                 float* __restrict__ out) {
  const int b = blockIdx.x;
  const int d = threadIdx.x;                 // 128 threads
  const float* zb = z + (size_t)b * N_A * D_H;
  float s = 0.f;
  for (int n = 0; n < N_A; ++n) s += zb[n * D_H + d];
  out[ZG_OFF + (size_t)b * D_H + d] = s;
  for (int i = d; i < N_A * D_H; i += D_H)
    out[ZO_OFF + (size_t)b * N_A * D_H + i] = zb[i];
}

// ---------------- atom_feat = z @ W_af + b_af ----------------
__global__ void atom_feat_kernel(const float* __restrict__ z,
                                 const float* __restrict__ W_af,
                                 const float* __restrict__ b_af,
                                 float* __restrict__ out) {
  int i = blockIdx.x * blockDim.x + threadIdx.x;
  const int total = B_G * N_A * A_DIM;
  if (i >= total) return;
  int row = i / A_DIM;
  int col = i - row * A_DIM;
  const float* zr = z + (size_t)row * D_H;
  float acc = b_af[col];
  for (int d = 0; d < D_H; ++d) acc = fmaf(zr[d], W_af[d * A_DIM + col], acc);
  out[AF_OFF + i] = acc;
}

// ---------------- sg_pred = log_softmax(z_G @ W_sg + b_sg) ----------------
__global__ void sg_kernel(const float* __restrict__ W_sg,
                          const float* __restrict__ b_sg,
                          float* __restrict__ out) {
  __shared__ float zg[D_H];
  __shared__ float lg[256];
  __shared__ float red[256];
  const int b = blockIdx.x;
  const int t = threadIdx.x;                 // 256 threads
  if (t < D_H) zg[t] = out[ZG_OFF + (size_t)b * D_H + t];
  __syncthreads();
  float v = -3.4e38f;
  if (t < SG_N) {
    float acc = b_sg[t];
    for (int d = 0; d < D_H; ++d) acc = fmaf(zg[d], W_sg[d * SG_N + t], acc);
    lg[t] = acc;
    v = acc;
  }
  red[t] = v;
  __syncthreads();
  for (int s = 128; s > 0; s >>= 1) {
    if (t < s) red[t] = fmaxf(red[t], red[t + s]);
    __syncthreads();
  }
  const float m = red[0];
  __syncthreads();
  red[t] = (t < SG_N) ? expf(lg[t] - m) : 0.f;
  __syncthreads();
  for (int s = 128; s > 0; s >>= 1) {
    if (t < s) red[t] += red[t + s];
    __syncthreads();
  }
  const float lse = m + logf(red[0]);
  if (t < SG_N) out[SGP_OFF + (size_t)b * SG_N + t] = lg[t] - lse;
}

// ---------------- fused bilinear + fc1 + log_softmax (WMMA) ----------------
// grid = B graphs, block = 256 (8 waves); wave w handles k = w, w+8, ...
// LDS layout (floats): Zs[32][132] | Tbuf[8][32][34] | Sc[400][65]
#define ZS_STRIDE 132
#define TW_STRIDE 34
#define SC_STRIDE 65
#define ZS_FLOATS (32 * ZS_STRIDE)
#define TW_FLOATS (32 * TW_STRIDE)
#define SC_FLOATS (N_A * N_A * SC_STRIDE)
#define BIL_SMEM_FLOATS (ZS_FLOATS + 8 * TW_FLOATS + SC_FLOATS)

__global__ void bilinear_kernel(const float* __restrict__ z,
                                const float* __restrict__ Wb,
                                const float* __restrict__ bb,
                                const float* __restrict__ W1,
                                const float* __restrict__ b1,
                                float* __restrict__ out) {
  extern __shared__ float smem[];
  float* Zs = smem;                            // 32 x 132 (rows >=20 zero)
  float* Tb = smem + ZS_FLOATS;                // 8 waves x 32 x 34
  float* Sc = Tb + 8 * TW_FLOATS;              // 400 x 65 (K padded)
  const int b    = blockIdx.x;
  const int tid  = threadIdx.x;
  const int wave = tid >> 5;
  const int lane = tid & 31;
  const int half = lane >> 4;                  // 0 or 1
  const int lrow = lane & 15;
  float* Tw = Tb + wave * TW_FLOATS;

  // stage Z (zero-padded to 32 rows) into LDS
  for (int i = tid; i < ZS_FLOATS; i += 256) {
    int r = i / ZS_STRIDE, c = i - r * ZS_STRIDE;
    float vv = 0.f;
    if (r < N_A && c < D_H) vv = z[(size_t)(b * N_A + r) * D_H + c];
    Zs[i] = vv;
  }
  __syncthreads();

  const v8f vzero = {0.f, 0.f, 0.f, 0.f, 0.f, 0.f, 0.f, 0.f};

  for (int ki = 0; ki < 8; ++ki) {
    const int k = wave + 8 * ki;
    const float* Wbk = Wb + (size_t)k * D_H * D_H;
    v8f Sacc[2][2];
    Sacc[0][0] = vzero; Sacc[0][1] = vzero;
    Sacc[1][0] = vzero; Sacc[1][1] = vzero;

    for (int ec = 0; ec < 4; ++ec) {           // 32-wide e-chunks of T
      v8f Tacc[2][2];
      Tacc[0][0] = vzero; Tacc[0][1] = vzero;
      Tacc[1][0] = vzero; Tacc[1][1] = vzero;

      // T[:, ec*32 .. ec*32+31] = Zpad(32x128) @ Wb[k][:, chunk]
      for (int s = 0; s < 32; ++s) {           // K-steps of 4 over D=128
        const int dcol = 4 * s + 2 * half;
        v2f a0 = *(const v2f*)&Zs[lrow * ZS_STRIDE + dcol];
        v2f a1 = *(const v2f*)&Zs[(16 + lrow) * ZS_STRIDE + dcol];
#pragma unroll
        for (int nt = 0; nt < 2; ++nt) {
          const int col = ec * 32 + nt * 16 + lrow;
          v2f bf;
          bf.x = Wbk[(size_t)dcol * D_H + col];
          bf.y = Wbk[(size_t)(dcol + 1) * D_H + col];
          Tacc[0][nt] = wmma4(a0, bf, Tacc[0][nt]);
          Tacc[1][nt] = wmma4(a1, bf, Tacc[1][nt]);
        }
      }
      // spill T chunk (C/D layout -> row major) to this wave's LDS buffer
#pragma unroll
      for (int mt = 0; mt < 2; ++mt)
#pragma unroll
        for (int nt = 0; nt < 2; ++nt)
#pragma unroll
          for (int r = 0; r < 8; ++r) {
            const int row = 16 * mt + r + 8 * half;
            const int col = 16 * nt + lrow;
            Tw[row * TW_STRIDE + col] = Tacc[mt][nt][r];
          }
      // S += Tchunk(32x32) @ Z[:, chunk]^T
      for (int s = 0; s < 8; ++s) {
        const int ecol = 4 * s + 2 * half;
        v2f a0 = *(const v2f*)&Tw[lrow * TW_STRIDE + ecol];
        v2f a1 = *(const v2f*)&Tw[(16 + lrow) * TW_STRIDE + ecol];
#pragma unroll
        for (int nt = 0; nt < 2; ++nt) {
          v2f bf = *(const v2f*)&Zs[(16 * nt + lrow) * ZS_STRIDE + ec * 32 + ecol];
          Sacc[0][nt] = wmma4(a0, bf, Sacc[0][nt]);
          Sacc[1][nt] = wmma4(a1, bf, Sacc[1][nt]);
        }
      }
    }

    const float bbk = bb[k];
#pragma unroll
    for (int mt = 0; mt < 2; ++mt)
#pragma unroll
      for (int nt = 0; nt < 2; ++nt)
#pragma unroll
        for (int r = 0; r < 8; ++r) {
          const int row = 16 * mt + r + 8 * half;
          const int col = 16 * nt + lrow;
          if (row < N_A && col < N_A)
            Sc[(row * N_A + col) * SC_STRIDE + k] = Sacc[mt][nt][r] + bbk;
        }
  }
  __syncthreads();

  // reuse Tbuf region for W1 (64x64) and b1
  float* W1s = Tb;
  float* b1s = Tb + K_E * K_E;
  for (int i = tid; i < K_E * K_E; i += 256) W1s[i] = W1[i];
  if (tid < K_E) b1s[tid] = b1[tid];
  __syncthreads();

  // fc1 + log_softmax over K per (n,m) pair
  for (int p = tid; p < N_A * N_A; p += 256) {
    float lgv[K_E];
#pragma unroll
    for (int j = 0; j < K_E; ++j) lgv[j] = b1s[j];
    for (int kk = 0; kk < K_E; ++kk) {
      const float sck = Sc[p * SC_STRIDE + kk];
#pragma unroll
      for (int j = 0; j < K_E; ++j) lgv[j] = fmaf(sck, W1s[kk * K_E + j], lgv[j]);
    }
    float m = -3.4e38f;
#pragma unroll
    for (int j = 0; j < K_E; ++j) m = fmaxf(m, lgv[j]);
    float ssum = 0.f;
#pragma unroll
    for (int j = 0; j < K_E; ++j) ssum += expf(lgv[j] - m);
    const float lse = m + logf(ssum);
    float* op = out + EP_OFF + ((size_t)b * N_A * N_A + p) * K_E;
#pragma unroll
    for (int j = 0; j < K_E; ++j) op[j] = lgv[j] - lse;
  }
}

// ---------------- dense adjacency ----------------
__global__ void adj_zero_kernel(float* __restrict__ out) {
  int i = blockIdx.x * blockDim.x + threadIdx.x;
  if (i < B_G * N_A * N_A) out[ADJ_OFF + i] = 0.f;
}

__global__ void adj_scatter_kernel(const int* __restrict__ ei, int E,
                                   float* __restrict__ out) {
  int e = blockIdx.x * blockDim.x + threadIdx.x;
  if (e >= E) return;
  const int src = ei[e];
  const int dst = ei[E + e];
  const int g = src / N_A;
  const int ls = src - g * N_A;
  const int ld = dst - g * N_A;
  atomicAdd(&out[ADJ_OFF + (size_t)g * N_A * N_A + ls * N_A + ld], 1.0f);
}

// ---------------- launcher ----------------
extern "C" void kernel_launch(void* const* d_in, const int* in_sizes, int n_in,
                              void* d_out, int out_size, void* d_ws, size_t ws_size,
                              hipStream_t stream) {
  const float* z    = (const float*)d_in[0];
  const float* W_af = (const float*)d_in[1];
  const float* b_af = (const float*)d_in[2];
  const float* Wb   = (const float*)d_in[3];
  const float* bb   = (const float*)d_in[4];
  const float* W1   = (const float*)d_in[5];
  const float* b1   = (const float*)d_in[6];
  const float* W_sg = (const float*)d_in[7];
  const float* b_sg = (const float*)d_in[8];
  const int*   ei   = (const int*)d_in[9];
  // d_in[10] = batch (implicit: batch[i] = i / N_A)
  float* out = (float*)d_out;
  const int E = in_sizes[9] / 2;

  // z_G + z passthrough
  pool_copy_kernel<<<B_G, D_H, 0, stream>>>(z, out);
  // atom features
  {
    const int total = B_G * N_A * A_DIM;
    atom_feat_kernel<<<(total + 255) / 256, 256, 0, stream>>>(z, W_af, b_af, out);
  }
  // space-group head (consumes z_G written by pool_copy_kernel)
  sg_kernel<<<B_G, 256, 0, stream>>>(W_sg, b_sg, out);
  // fused bilinear + fc1 + log_softmax (WMMA)
  bilinear_kernel<<<B_G, 256, BIL_SMEM_FLOATS * sizeof(float), stream>>>(
      z, Wb, bb, W1, b1, out);
  // dense adjacency
  adj_zero_kernel<<<(B_G * N_A * N_A + 255) / 256, 256, 0, stream>>>(out);
  adj_scatter_kernel<<<(E + 255) / 256, 256, 0, stream>>>(ei, E, out);
}